// GCNModel_9113920602300
// MI455X (gfx1250) — compile-verified
//
#include <hip/hip_runtime.h>
#include <hip/hip_bf16.h>

typedef float v2f __attribute__((ext_vector_type(2)));
typedef float v8f __attribute__((ext_vector_type(8)));

#define KT  32   // K-strip per LDS stage
#define KTP 34   // padded stride (even -> 8B-aligned v2f loads, spreads banks)

// ---------------- degree / norm prep ----------------
__global__ void zero_f32(float* p, int n) {
    int i = blockIdx.x * blockDim.x + threadIdx.x;
    if (i < n) p[i] = 0.0f;
}

__global__ void deg_accum(const int* __restrict__ dst, const float* __restrict__ w,
                          float* __restrict__ deg, int e) {
    int i = blockIdx.x * blockDim.x + threadIdx.x;
    if (i < e) atomicAdd(&deg[dst[i]], w[i]);
}

__global__ void deg_finalize(float* __restrict__ d, int n) {
    int i = blockIdx.x * blockDim.x + threadIdx.x;
    if (i < n) d[i] = rsqrtf(d[i] + 1.0f);   // +1 = self-loop weight; always > 0
}

__global__ void edge_norm(const int* __restrict__ src, const int* __restrict__ dst,
                          const float* __restrict__ w, const float* __restrict__ dinv,
                          float* __restrict__ nrm, int e) {
    int i = blockIdx.x * blockDim.x + threadIdx.x;
    if (i < e) nrm[i] = dinv[src[i]] * w[i] * dinv[dst[i]];
}

// ---------------- WMMA f32 GEMM: H[nrows x ncols] = act(X)[nrows x K] @ W[K x ncols] ----------------
// block: 256 threads = 8 waves; block tile 128(M) x 64(N); wave w -> rows w*16..+15, 4 col tiles.
__global__ __launch_bounds__(256) void gemm_wmma_f32(
    const float* __restrict__ X, const float* __restrict__ Wm,
    float* __restrict__ H, int nrows, int K, int ncols, int relu_in)
{
    __shared__ float As[128 * KTP];   // [row][k]
    __shared__ float Bs[64 * KTP];    // [col][k]  (transposed)

    const int tid  = threadIdx.x;
    const int lane = tid & 31;
    const int wv   = tid >> 5;
    const int rowBase = blockIdx.x * 128;
    const int colBase = blockIdx.y * 64;
    const int lm = lane & 15;            // M/N index within tile
    const int kh = (lane >> 4) * 2;      // K half offset (0 or 2)

    v8f acc[4] = {};

    for (int kt = 0; kt < K; kt += KT) {
        // stage A strip 128 x KT (coalesced along k), fuse ReLU of previous layer
        for (int i = tid; i < 128 * KT; i += 256) {
            int r = i >> 5, k = i & (KT - 1);
            int gr = rowBase + r; if (gr >= nrows) gr = nrows - 1;   // clamp, store-guarded
            float v = X[(long)gr * K + kt + k];
            if (relu_in) v = fmaxf(v, 0.0f);
            As[r * KTP + k] = v;
        }
        // stage B strip KT x 64, transposed into [col][k]
        for (int i = tid; i < KT * 64; i += 256) {
            int k = i >> 6, c = i & 63;
            Bs[c * KTP + k] = Wm[(long)(kt + k) * ncols + colBase + c];
        }
        __syncthreads();

        const float* arow = &As[(wv * 16 + lm) * KTP];
        #pragma unroll
        for (int k0 = 0; k0 < KT; k0 += 4) {
            v2f a = *(const v2f*)&arow[k0 + kh];               // A[m][k0+2h+{0,1}]
            #pragma unroll
            for (int c = 0; c < 4; ++c) {
                v2f b = *(const v2f*)&Bs[(c * 16 + lm) * KTP + k0 + kh]; // B[k0+2h+{0,1}][n]
                acc[c] = __builtin_amdgcn_wmma_f32_16x16x4_f32(
                    false, a, false, b, (short)0, acc[c], false, false);
            }
        }
        __syncthreads();
    }

    // C/D layout: vgpr r holds (M = r + 8*(lane>=16), N = lane%16)
    const int mh = (lane >> 4) * 8;
    #pragma unroll
    for (int c = 0; c < 4; ++c) {
        #pragma unroll
        for (int r = 0; r < 8; ++r) {
            int grow = rowBase + wv * 16 + r + mh;
            if (grow < nrows)
                H[(long)grow * ncols + colBase + c * 16 + lm] = acc[c][r];
        }
    }
}

// ---------------- aggregation ----------------
// OUT[i][f] = bias[f] + dinv[i]^2 * H[i][f]   (self-loop term; full overwrite)
__global__ void agg_init(const float* __restrict__ H, const float* __restrict__ bias,
                         const float* __restrict__ dinv, float* __restrict__ OUT,
                         long total, int shift) {
    long idx = (long)blockIdx.x * blockDim.x + threadIdx.x;
    if (idx >= total) return;
    long node = idx >> shift;
    int  f4   = (int)(idx & ((1 << shift) - 1));
    float di = dinv[node]; float s = di * di;
    const float4 hv = *(const float4*)&H[(node << shift << 2) + f4 * 4];
    const float4 bv = *(const float4*)&bias[f4 * 4];
    float4 o; o.x = bv.x + s * hv.x; o.y = bv.y + s * hv.y;
    o.z = bv.z + s * hv.z; o.w = bv.w + s * hv.w;
    *(float4*)&OUT[(node << shift << 2) + f4 * 4] = o;
}

// OUT[dst[e]][f] += nrm[e] * H[src[e]][f] ;  (F/4) threads per edge, float4 per thread
__global__ void scatter_add(const float* __restrict__ H, const int* __restrict__ src,
                            const int* __restrict__ dst, const float* __restrict__ nrm,
                            float* __restrict__ OUT, long total, int shift) {
    long idx = (long)blockIdx.x * blockDim.x + threadIdx.x;
    if (idx >= total) return;
    long e  = idx >> shift;
    int  t  = (int)(idx & ((1 << shift) - 1));
    int  s  = src[e], d = dst[e];
    float n = nrm[e];
    const float4 hv = *(const float4*)&H[((long)s << shift << 2) + t * 4];
    float* o = &OUT[((long)d << shift << 2) + t * 4];
    atomicAdd(o + 0, n * hv.x);
    atomicAdd(o + 1, n * hv.y);
    atomicAdd(o + 2, n * hv.z);
    atomicAdd(o + 3, n * hv.w);
}

// ---------------- launch ----------------
extern "C" void kernel_launch(void* const* d_in, const int* in_sizes, int n_in,
                              void* d_out, int out_size, void* d_ws, size_t ws_size,
                              hipStream_t stream) {
    const float* x   = (const float*)d_in[0];
    const int*   ei  = (const int*)d_in[1];
    const float* ew  = (const float*)d_in[2];
    const float* W0  = (const float*)d_in[3];
    const float* b0  = (const float*)d_in[4];
    const float* W1  = (const float*)d_in[5];
    const float* b1  = (const float*)d_in[6];
    const float* W2  = (const float*)d_in[7];
    const float* b2  = (const float*)d_in[8];
    float* out = (float*)d_out;

    const int N = in_sizes[0] / 128;   // 100000
    const int E = in_sizes[2];         // 1600000
    const int FIN = 128, HID = 256, NCLS = 64;
    const int* srcp = ei;
    const int* dstp = ei + E;

    float* dinv   = (float*)d_ws;
    float* nrm    = dinv + N;
    float* hbuf   = nrm + E;
    float* aggbuf = hbuf + (size_t)N * HID;

    const int B = 256;
    dim3 blk(B);

    // normalization
    zero_f32<<<(N + B - 1) / B, blk, 0, stream>>>(dinv, N);
    deg_accum<<<(E + B - 1) / B, blk, 0, stream>>>(dstp, ew, dinv, E);
    deg_finalize<<<(N + B - 1) / B, blk, 0, stream>>>(dinv, N);
    edge_norm<<<(E + B - 1) / B, blk, 0, stream>>>(srcp, dstp, ew, dinv, nrm, E);

    const int mblocks = (N + 127) / 128;
    long t256 = (long)N * (HID / 4);           // agg_init work, F=256
    long s256 = (long)E << 6;                  // scatter work, F=256
    long t64  = (long)N * (NCLS / 4);
    long s64  = (long)E << 4;

    // layer 1: x(128) -> 256, relu later fused into next gemm's load
    gemm_wmma_f32<<<dim3(mblocks, HID / 64), blk, 0, stream>>>(x, W0, hbuf, N, FIN, HID, 0);
    agg_init<<<(unsigned)((t256 + B - 1) / B), blk, 0, stream>>>(hbuf, b0, dinv, aggbuf, t256, 6);
    scatter_add<<<(unsigned)((s256 + B - 1) / B), blk, 0, stream>>>(hbuf, srcp, dstp, nrm, aggbuf, s256, 6);

    // layer 2: relu(agg) @ W1 -> 256
    gemm_wmma_f32<<<dim3(mblocks, HID / 64), blk, 0, stream>>>(aggbuf, W1, hbuf, N, HID, HID, 1);
    agg_init<<<(unsigned)((t256 + B - 1) / B), blk, 0, stream>>>(hbuf, b1, dinv, aggbuf, t256, 6);
    scatter_add<<<(unsigned)((s256 + B - 1) / B), blk, 0, stream>>>(hbuf, srcp, dstp, nrm, aggbuf, s256, 6);

    // layer 3: relu(agg) @ W2 -> 64, aggregate straight into d_out (no relu)
    gemm_wmma_f32<<<dim3(mblocks, NCLS / 64), blk, 0, stream>>>(aggbuf, W2, hbuf, N, HID, NCLS, 1);
    agg_init<<<(unsigned)((t64 + B - 1) / B), blk, 0, stream>>>(hbuf, b2, dinv, out, t64, 4);
    scatter_add<<<(unsigned)((s64 + B - 1) / B), blk, 0, stream>>>(hbuf, srcp, dstp, nrm, out, s64, 4);
}